// POELayer_88055419502920
// MI455X (gfx1250) — compile-verified
//
#include <hip/hip_runtime.h>
#include <math.h>

typedef __attribute__((ext_vector_type(2))) float v2f;
typedef __attribute__((ext_vector_type(8))) float v8f;

#define EPSF 1e-12f

// One thread = one batch element. 16-joint chain kept entirely in registers.
// Epilogue: per-wave WMMA (v_wmma_f32_16x16x4_f32) applies the constant T_init
// to all 32 accumulated 4x4 transforms, staged through LDS.
__global__ __launch_bounds__(256) void poe_fwd_kernel(
    const float* __restrict__ q,        // (B,16)
    const float* __restrict__ twist,    // (16,6)
    const float* __restrict__ initp,    // (3)
    const float* __restrict__ initrpy,  // (3)
    float* __restrict__ outT,           // (B,16)  row-major 4x4
    float* __restrict__ twl)            // (B,96)  (B,16,6)
{
    __shared__ float smem[256 * 16];    // 16 KB: 8 waves x 32 lanes x 16 floats
    const int tid  = blockIdx.x * 256 + threadIdx.x;
    const int lane = threadIdx.x & 31;
    float* ws = &smem[(threadIdx.x >> 5) * 512];

    // Accumulated transform T = [R p; 0 0 0 1], starts at identity.
    float R00 = 1.f, R01 = 0.f, R02 = 0.f;
    float R10 = 0.f, R11 = 1.f, R12 = 0.f;
    float R20 = 0.f, R21 = 0.f, R22 = 1.f;
    float p0 = 0.f, p1 = 0.f, p2 = 0.f;

    const float* qp   = q   + (size_t)tid * 16;
    float*       tout = twl + (size_t)tid * 96;

    #pragma unroll
    for (int j = 0; j < 16; ++j) {
        // wave-uniform twist -> scalar loads
        const float wx = twist[j*6+0], wy = twist[j*6+1], wz = twist[j*6+2];
        const float vx = twist[j*6+3], vy = twist[j*6+4], vz = twist[j*6+5];

        // ---- local twist from T BEFORE this joint ----
        // w_loc = R^T w ; v_loc = R^T (v - p x w)
        const float c0 = p1*wz - p2*wy;
        const float c1 = p2*wx - p0*wz;
        const float c2 = p0*wy - p1*wx;
        const float m0 = vx - c0, m1 = vy - c1, m2 = vz - c2;
        const float wl0 = R00*wx + R10*wy + R20*wz;
        const float wl1 = R01*wx + R11*wy + R21*wz;
        const float wl2 = R02*wx + R12*wy + R22*wz;
        const float vl0 = R00*m0 + R10*m1 + R20*m2;
        const float vl1 = R01*m0 + R11*m1 + R21*m2;
        const float vl2 = R02*m0 + R12*m1 + R22*m2;
        float2* t2 = (float2*)(tout + j*6);      // 8B aligned (24*j offset)
        t2[0] = make_float2(wl0, wl1);
        t2[1] = make_float2(wl2, vl0);
        t2[2] = make_float2(vl1, vl2);

        // ---- srodrigues: S = [SR sp; 0 0 0 1] ----
        const float th  = sqrtf(wx*wx + wy*wy + wz*wz) + EPSF;
        const float inv = 1.0f / th;
        const float ux = wx*inv, uy = wy*inv, uz = wz*inv;
        const float nx = vx*inv, ny = vy*inv, nz = vz*inv;
        const float qt = qp[j] * th;
        const float s  = __sinf(qt);
        const float cc = 1.0f - __cosf(qt);
        const float n2 = ux*ux + uy*uy + uz*uz;
        // W2 = u u^T - |u|^2 I  (skew identity)
        const float A00 = ux*ux - n2, A11 = uy*uy - n2, A22 = uz*uz - n2;
        const float A01 = ux*uy, A02 = ux*uz, A12 = uy*uz;
        const float S00 = 1.0f + cc*A00;
        const float S01 = cc*A01 - s*uz;
        const float S02 = cc*A02 + s*uy;
        const float S10 = cc*A01 + s*uz;
        const float S11 = 1.0f + cc*A11;
        const float S12 = cc*A12 - s*ux;
        const float S20 = cc*A02 - s*uy;
        const float S21 = cc*A12 + s*ux;
        const float S22 = 1.0f + cc*A22;
        // sp = t*vn + cc*(u x vn) + (t-s)*(u(u.vn) - |u|^2 vn)
        const float Wn0 = uy*nz - uz*ny;
        const float Wn1 = uz*nx - ux*nz;
        const float Wn2 = ux*ny - uy*nx;
        const float udn = ux*nx + uy*ny + uz*nz;
        const float Qn0 = ux*udn - n2*nx;
        const float Qn1 = uy*udn - n2*ny;
        const float Qn2 = uz*udn - n2*nz;
        const float tsd = qt - s;
        const float sp0 = qt*nx + cc*Wn0 + tsd*Qn0;
        const float sp1 = qt*ny + cc*Wn1 + tsd*Qn1;
        const float sp2 = qt*nz + cc*Wn2 + tsd*Qn2;

        // ---- T = T * S (affine) ----
        const float N00 = R00*S00 + R01*S10 + R02*S20;
        const float N01 = R00*S01 + R01*S11 + R02*S21;
        const float N02 = R00*S02 + R01*S12 + R02*S22;
        const float N10 = R10*S00 + R11*S10 + R12*S20;
        const float N11 = R10*S01 + R11*S11 + R12*S21;
        const float N12 = R10*S02 + R11*S12 + R12*S22;
        const float N20 = R20*S00 + R21*S10 + R22*S20;
        const float N21 = R20*S01 + R21*S11 + R22*S21;
        const float N22 = R20*S02 + R21*S12 + R22*S22;
        const float np0 = R00*sp0 + R01*sp1 + R02*sp2 + p0;
        const float np1 = R10*sp0 + R11*sp1 + R12*sp2 + p1;
        const float np2 = R20*sp0 + R21*sp1 + R22*sp2 + p2;
        R00=N00; R01=N01; R02=N02;
        R10=N10; R11=N11; R12=N12;
        R20=N20; R21=N21; R22=N22;
        p0=np0; p1=np1; p2=np2;
    }

    // ---- stage the 32 accumulated 4x4 matrices of this wave into LDS ----
    float4* wsv = (float4*)ws;
    wsv[lane*4+0] = make_float4(R00, R01, R02, p0);
    wsv[lane*4+1] = make_float4(R10, R11, R12, p1);
    wsv[lane*4+2] = make_float4(R20, R21, R22, p2);
    wsv[lane*4+3] = make_float4(0.f, 0.f, 0.f, 1.f);
    __builtin_amdgcn_wave_barrier();

    // ---- T_init = pr2t(init_p, init_rpy) ----
    const float rr = initrpy[0], rp = initrpy[1], ry = initrpy[2];
    const float crr=__cosf(rr), srr=__sinf(rr);
    const float crp=__cosf(rp), srp=__sinf(rp);
    const float cry=__cosf(ry), sry=__sinf(ry);
    const float T00=cry*crp, T01=cry*srp*srr - sry*crr, T02=cry*srp*crr + sry*srr, T03=initp[0];
    const float T10=sry*crp, T11=sry*srp*srr + cry*crr, T12=sry*srp*crr - cry*srr, T13=initp[1];
    const float T20=-srp,    T21=crp*srr,               T22=crp*crr,               T23=initp[2];
    // row 3 = [0,0,0,1]

    // ---- B fragment: B[k][n] = T_init[k][n&3] (columns replicated 4x over N)
    // layout: VGPR v, lanes 0-15 -> K=v, lanes 16-31 -> K=v+2, N=lane%16
    const int  col = lane & 3;
    const bool hi  = (lane >= 16);
    const float a0 = hi ? T20 : T00, a1 = hi ? T21 : T01, a2 = hi ? T22 : T02, a3 = hi ? T23 : T03;
    const float b0 = hi ? 0.f : T10, b1 = hi ? 0.f : T11, b2 = hi ? 0.f : T12, b3 = hi ? 1.f : T13;
    v2f bfrag;
    bfrag.x = (col == 0) ? a0 : (col == 1) ? a1 : (col == 2) ? a2 : a3;
    bfrag.y = (col == 0) ? b0 : (col == 1) ? b1 : (col == 2) ? b2 : b3;

    // A layout (16x4 f32): lane -> M=lane%16; lo half K=0,1 ; hi half K=2,3
    const int m    = lane & 15;
    const int kb   = hi ? 2 : 0;
    const int src0 = (m >> 2) * 16 + (m & 3) * 4 + kb;

    #pragma unroll
    for (int g = 0; g < 8; ++g) {
        v2f afrag;
        afrag.x = ws[g*64 + src0];
        afrag.y = ws[g*64 + src0 + 1];
        v8f cacc = {0.f,0.f,0.f,0.f,0.f,0.f,0.f,0.f};
        v8f d = __builtin_amdgcn_wmma_f32_16x16x4_f32(
            false, afrag, false, bfrag, (short)0, cacc, false, false);
        // D layout: VGPR v, lane l -> M = v + 8*(l>=16), N = l%16.
        // Columns replicate mod 4, so lanes 0-3 / 16-19 hold all results.
        if ((lane & 12) == 0) {
            const int half8 = hi ? 8 : 0;
            #pragma unroll
            for (int v = 0; v < 8; ++v) {
                const int M = v + half8;             // row index within the 16
                ws[(g*4 + (M >> 2))*16 + (M & 3)*4 + col] = d[v];
            }
        }
    }
    __builtin_amdgcn_wave_barrier();

    // ---- write final transforms (B128 x 4 per thread) ----
    float4* og = (float4*)(outT + (size_t)tid * 16);
    og[0] = wsv[lane*4 + 0];
    og[1] = wsv[lane*4 + 1];
    og[2] = wsv[lane*4 + 2];
    og[3] = wsv[lane*4 + 3];
}

extern "C" void kernel_launch(void* const* d_in, const int* in_sizes, int n_in,
                              void* d_out, int out_size, void* d_ws, size_t ws_size,
                              hipStream_t stream) {
    (void)n_in; (void)out_size; (void)d_ws; (void)ws_size;
    const float* q       = (const float*)d_in[0];
    const float* twist   = (const float*)d_in[1];
    const float* initp   = (const float*)d_in[2];
    const float* initrpy = (const float*)d_in[3];
    const int B = in_sizes[0] / 16;          // 262144
    float* outT = (float*)d_out;             // B*16 floats
    float* twl  = outT + (size_t)B * 16;     // B*96 floats
    const int nblk = B / 256;                // exact: 1024 blocks of 256
    poe_fwd_kernel<<<nblk, 256, 0, stream>>>(q, twist, initp, initrpy, outT, twl);
}